// TensorProductMultiLayerPerceptron_60782377173405
// MI455X (gfx1250) — compile-verified
//
#include <hip/hip_runtime.h>
#include <math.h>

typedef __attribute__((ext_vector_type(2))) float v2f;
typedef __attribute__((ext_vector_type(8))) float v8f;
typedef __attribute__((ext_vector_type(4))) int v4i;

#define NB 128   // batch
#define MD 64    // MUL = EMB = 64

#if __has_builtin(__builtin_amdgcn_global_load_async_to_lds_b128)
#define ASYNC_STAGE 1
typedef __attribute__((address_space(1))) v4i gv4i;
typedef __attribute__((address_space(3))) v4i lv4i;
#else
#define ASYNC_STAGE 0
#endif

// ---------------- MLP layer: Y = gelu_tanh(X@W * scale)/PHI * out_scale ----------------
__global__ void k_mlp(const float* __restrict__ X, const float* __restrict__ W,
                      float* __restrict__ Y, int K, int N,
                      float scale, float phi_inv, float out_scale)
{
    int idx = blockIdx.x * blockDim.x + threadIdx.x;
    if (idx >= NB * N) return;
    int b = idx / N, n = idx - b * N;
    const float* xr = X + (size_t)b * K;
    const float* wc = W + n;
    float acc = 0.f;
    for (int k = 0; k < K; ++k) acc = fmaf(xr[k], wc[(size_t)k * N], acc);
    float x = acc * scale;
    float t = tanhf(0.7978845608028654f * (x + 0.044715f * x * x * x));
    Y[idx] = 0.5f * x * (1.f + t) * phi_inv * out_scale;
}

__global__ void k_zero(float* __restrict__ p, int n)
{
    int i = blockIdx.x * blockDim.x + threadIdx.x;
    if (i < n) p[i] = 0.f;
}

// ---------------- Main tensor-product kernel ----------------
// grid: 640 blocks = 5 P-tensors x 64 u x 2 v-halves ; 256 threads = 8 waves.
// Per slab (u,v): Wt[b,w] = sum_x hs[b,x] * P[x,u,v,w]  (M=128,N=64,K=64 fp32 WMMA)
// then acc_c[b,w] += coef_c(b,u,v) * Wt[b,w].
__global__ __launch_bounds__(256) void k_tp(
    const float* __restrict__ P0c, const float* __restrict__ P1c,
    const float* __restrict__ P2c, const float* __restrict__ P3c,
    const float* __restrict__ P4c,
    const float* __restrict__ x1, const float* __restrict__ x2,
    const float* __restrict__ hsg,
    float* __restrict__ acc0, float* __restrict__ acc1, float* __restrict__ acc2)
{
    __shared__ float slab[64 * 64];   // 16KB: P[x][w] slab for current (u,v)
    __shared__ float uside[128 * 4];  // s1[b,u], v1[b,u,0..2]
    __shared__ float vside[128 * 4];  // s2[b,v], v2[b,v,0..2]
    __shared__ float coefs[3 * 128];  // coef[c][b] for current (u,v)

    const int bx  = blockIdx.x;
    const int p   = bx >> 7;           // 0..4
    const int rem = bx & 127;
    const int u   = rem >> 1;          // 0..63
    const int v0  = (rem & 1) * 32;    // v half

    const float* Pt = (p == 0) ? P0c : (p == 1) ? P1c : (p == 2) ? P2c : (p == 3) ? P3c : P4c;

    const int tid  = threadIdx.x;
    const int lane = tid & 31;
    const int wave = tid >> 5;         // 0..7 -> b-tile
    const int kh   = lane >> 4;        // lane half (K offset / M offset selector)
    const int ln   = lane & 15;
    const int b0   = wave * 16;

    // ---- A operand (hs b-tile) resident in registers: 16 k-steps x 2 regs ----
    float a0[16], a1[16];
    {
        const float* ar = hsg + (size_t)(b0 + ln) * 64;
#pragma unroll
        for (int t = 0; t < 16; ++t) {
            a0[t] = ar[4 * t + 2 * kh];
            a1[t] = ar[4 * t + 1 + 2 * kh];
        }
    }

    if (tid < 128) {
        int b = tid;
        uside[b * 4 + 0] = x1[b * 256 + u];
        uside[b * 4 + 1] = x1[b * 256 + 64 + 3 * u + 0];
        uside[b * 4 + 2] = x1[b * 256 + 64 + 3 * u + 1];
        uside[b * 4 + 3] = x1[b * 256 + 64 + 3 * u + 2];
    }

    v8f accO[3][4];
#pragma unroll
    for (int c = 0; c < 3; ++c)
#pragma unroll
        for (int nt = 0; nt < 4; ++nt)
#pragma unroll
            for (int r = 0; r < 8; ++r) accO[c][nt][r] = 0.f;

    const float inv_sq3 = 0.5773502691896258f;

    for (int s = 0; s < 32; ++s) {
        const int v = v0 + s;
        const float* src = Pt + (size_t)u * 4096 + (size_t)v * 64; // P[x=0][u][v][w=0]

        __syncthreads(); // previous compute done reading slab/coefs

        // stage slab: 64 rows (x) x 64 floats, row stride 262144 floats
#if ASYNC_STAGE
#pragma unroll
        for (int i = 0; i < 4; ++i) {
            int idx = tid + i * 256;
            int xr  = idx >> 4;
            int c4  = idx & 15;
            __builtin_amdgcn_global_load_async_to_lds_b128(
                (gv4i*)(src + (size_t)xr * 262144 + c4 * 4),
                (lv4i*)(&slab[xr * 64 + c4 * 4]),
                0, 0);
        }
#else
#pragma unroll
        for (int i = 0; i < 4; ++i) {
            int idx = tid + i * 256;
            int xr  = idx >> 4;
            int c4  = idx & 15;
            float4 val = *(const float4*)(src + (size_t)xr * 262144 + c4 * 4);
            *(float4*)(&slab[xr * 64 + c4 * 4]) = val;
        }
#endif
        if (tid < 128) {
            int b = tid;
            vside[b * 4 + 0] = x2[b * 256 + v];
            vside[b * 4 + 1] = x2[b * 256 + 64 + 3 * v + 0];
            vside[b * 4 + 2] = x2[b * 256 + 64 + 3 * v + 1];
            vside[b * 4 + 3] = x2[b * 256 + 64 + 3 * v + 2];
        }
#if ASYNC_STAGE
#if __has_builtin(__builtin_amdgcn_s_wait_asynccnt)
        __builtin_amdgcn_s_wait_asynccnt(0);
#else
        asm volatile("s_wait_asynccnt 0x0" ::: "memory");
#endif
#endif
        __syncthreads();

        // per-b coefficients for this (u,v)
        if (tid < 128) {
            int b = tid;
            float us0 = uside[b * 4 + 0], us1 = uside[b * 4 + 1];
            float us2 = uside[b * 4 + 2], us3 = uside[b * 4 + 3];
            float vs0 = vside[b * 4 + 0], vs1 = vside[b * 4 + 1];
            float vs2 = vside[b * 4 + 2], vs3 = vside[b * 4 + 3];
            float c0, c1, c2;
            if (p == 0)      { c0 = us0 * vs0; c1 = 0.f; c2 = 0.f; }
            else if (p == 1) { c0 = us0 * vs1; c1 = us0 * vs2; c2 = us0 * vs3; }
            else if (p == 2) { c0 = us1 * vs0; c1 = us2 * vs0; c2 = us3 * vs0; }
            else if (p == 3) { c0 = inv_sq3 * (us1 * vs1 + us2 * vs2 + us3 * vs3); c1 = 0.f; c2 = 0.f; }
            else             { c0 = us2 * vs3 - us3 * vs2;     // cross(v1,v2)
                               c1 = us3 * vs1 - us1 * vs3;
                               c2 = us1 * vs2 - us2 * vs1; }
            coefs[0 * 128 + b] = c0;
            coefs[1 * 128 + b] = c1;
            coefs[2 * 128 + b] = c2;
        }
        // prefetch next slab's rows into L2 while we compute
        if (tid < 128 && s + 1 < 32) {
            __builtin_prefetch(src + 64 + (size_t)(tid >> 1) * 262144 + (tid & 1) * 32, 0, 1);
        }
        __syncthreads();

        // coef regs for this wave's b-tile (C layout: M = r + 8*kh)
        float cf[3][8];
#pragma unroll
        for (int c = 0; c < 3; ++c)
#pragma unroll
            for (int r = 0; r < 8; ++r) cf[c][r] = coefs[c * 128 + b0 + 8 * kh + r];

#pragma unroll
        for (int nt = 0; nt < 4; ++nt) {
            const int col = nt * 16 + ln;
            v8f acc = {0.f, 0.f, 0.f, 0.f, 0.f, 0.f, 0.f, 0.f};
#pragma unroll
            for (int t = 0; t < 16; ++t) {
                v2f av = {a0[t], a1[t]};
                int rbase = (4 * t + 2 * kh) * 64 + col;
                v2f bv = {slab[rbase], slab[rbase + 64]};
                acc = __builtin_amdgcn_wmma_f32_16x16x4_f32(
                    false, av, false, bv, (short)0, acc, false, false);
            }
#pragma unroll
            for (int c = 0; c < 3; ++c)
#pragma unroll
                for (int r = 0; r < 8; ++r)
                    accO[c][nt][r] = fmaf(cf[c][r], acc[r], accO[c][nt][r]);
        }
    }

    // ---- cross-block accumulation ----
    float* dst;
    int stride, ncomp;
    if (p == 0 || p == 3) { dst = acc0; stride = 1; ncomp = 1; }
    else if (p == 4)      { dst = acc2; stride = 3; ncomp = 3; }
    else                  { dst = acc1; stride = 3; ncomp = 3; }

#pragma unroll
    for (int c = 0; c < 3; ++c) {
        if (c >= ncomp) break;
#pragma unroll
        for (int nt = 0; nt < 4; ++nt)
#pragma unroll
            for (int r = 0; r < 8; ++r) {
                int b = b0 + r + 8 * kh;
                int w = nt * 16 + ln;
                unsafeAtomicAdd(dst + (size_t)(b * 64 + w) * stride + c, accO[c][nt][r]);
            }
    }
}

// ---------------- final assembly: apply alphas, write 128x448 ----------------
__global__ void k_final(const float* __restrict__ a0, const float* __restrict__ a1,
                        const float* __restrict__ a2, float* __restrict__ out)
{
    int idx = blockIdx.x * blockDim.x + threadIdx.x;
    if (idx >= NB * 448) return;
    int b = idx / 448, q = idx - b * 448;
    const float A01 = 0.011048543456039806f; // 1/(64*sqrt(2)) == ALPHA_01 == ALPHA_2*INV_SQ2
    float val;
    if (q < 64)        val = A01 * a0[b * 64 + q];
    else if (q < 256)  val = A01 * a1[b * 192 + (q - 64)];   // (b,w,j) w-major
    else               val = A01 * a2[b * 192 + (q - 256)];
    out[idx] = val;
}

extern "C" void kernel_launch(void* const* d_in, const int* in_sizes, int n_in,
                              void* d_out, int out_size, void* d_ws, size_t ws_size,
                              hipStream_t stream)
{
    (void)in_sizes; (void)n_in; (void)out_size; (void)ws_size;
    const float* emb = (const float*)d_in[0];
    const float* x1  = (const float*)d_in[1];
    const float* x2  = (const float*)d_in[2];
    const float* W0  = (const float*)d_in[3];
    const float* W1  = (const float*)d_in[4];
    const float* W2  = (const float*)d_in[5];
    const float* P0  = (const float*)d_in[6];
    const float* P1  = (const float*)d_in[7];
    const float* P2  = (const float*)d_in[8];
    const float* P3  = (const float*)d_in[9];
    const float* P4  = (const float*)d_in[10];

    float* ws   = (float*)d_ws;
    float* h1   = ws;            // 128*256
    float* h2   = ws + 32768;    // 128*256
    float* hs   = ws + 65536;    // 128*64
    float* acc0 = ws + 73728;    // 128*64
    float* acc1 = ws + 81920;    // 128*64*3
    float* acc2 = ws + 106496;   // 128*64*3

    // PHI_C via trapz of tanh-approx gelu^2 * N(0,1), matching the reference
    double s = 0.0, fprev = 0.0;
    const double dz = 16.0 / 4000.0;
    for (int i = 0; i <= 4000; ++i) {
        double z = -8.0 + dz * i;
        double t = tanh(0.7978845608028654 * (z + 0.044715 * z * z * z));
        double g = 0.5 * z * (1.0 + t);
        double f = g * g * exp(-0.5 * z * z) * 0.3989422804014327;
        if (i > 0) s += 0.5 * (f + fprev) * dz;
        fprev = f;
    }
    float phi_inv = (float)(1.0 / sqrt(s));

    // MLP: scales are in_dim^-0.5 ; last layer also folds hs = h/8
    k_mlp<<<128, 256, 0, stream>>>(emb, W0, h1, 64, 256, 0.125f,  phi_inv, 1.f);
    k_mlp<<<128, 256, 0, stream>>>(h1,  W1, h2, 256, 256, 0.0625f, phi_inv, 1.f);
    k_mlp<<<32,  256, 0, stream>>>(h2,  W2, hs, 256, 64,  0.0625f, phi_inv, 0.125f);

    k_zero<<<(57344 + 255) / 256, 256, 0, stream>>>(acc0, 57344); // acc0+acc1+acc2 contiguous

    k_tp<<<640, 256, 0, stream>>>(P0, P1, P2, P3, P4, x1, x2, hs, acc0, acc1, acc2);

    k_final<<<(NB * 448 + 255) / 256, 256, 0, stream>>>(acc0, acc1, acc2, (float*)d_out);
}